// AdaptivePseudoLabelRefinement_48722109006326
// MI455X (gfx1250) — compile-verified
//
#include <hip/hip_runtime.h>
#include <hip/hip_bf16.h>
#include <math.h>

// Problem constants (from reference setup_inputs)
#define BANK 2048
#define DFEAT 131072          // 512*16*16
#define NB 8                  // B
#define NA 16                 // A
#define NC 19                 // C
#define HW 16384              // 128*128
#define KSEL 5
#define NSEG 8
#define KSEG (DFEAT / NSEG)   // 16384
#define NTILES (BANK / 16)    // 128

typedef float v2f __attribute__((ext_vector_type(2)));
typedef float v8f __attribute__((ext_vector_type(8)));

// ---------------------------------------------------------------------------
// Kernel 1: fused WMMA partial GEMM (dot(tgt_b, queue_n)) + queue row sumsq.
// One wave per (n-tile of 16 rows, K-segment). f32 WMMA 16x16x4:
//   A (16x4): lanes 0-15 hold (K0,K1) for M=lane; lanes 16-31 hold (K2,K3).
//   B (4x16): lanes 0-15 hold (K0,K1) for N=lane; lanes 16-31 hold (K2,K3).
//   C/D (16x16 f32): VGPR v, lanes 0-15 -> M=v (our target index b).
// ---------------------------------------------------------------------------
__global__ __launch_bounds__(32) void apr_gemm_partial(
    const float* __restrict__ queue, const float* __restrict__ tgt,
    float* __restrict__ dots, float* __restrict__ sqq)
{
    const int tile = blockIdx.x % NTILES;
    const int seg  = blockIdx.x / NTILES;
    const int lane = threadIdx.x;            // 0..31
    const int m    = lane & 15;              // matrix row within half
    const int half = lane >> 4;              // 0: K0/K1, 1: K2/K3
    const int nrow = (tile << 4) + m;        // bank row handled by this lane
    const float tscale = (m < 8) ? 1.0f : 0.0f;   // pad targets 8..15 with zero
    const float* __restrict__ qrow = queue + (size_t)nrow * DFEAT;
    const float* __restrict__ trow = tgt   + (size_t)(m & 7) * DFEAT;

    v8f acc = {};
    float sumsq = 0.0f;
    const int k0 = seg * KSEG;
    const int k1 = k0 + KSEG;
#pragma unroll 4
    for (int k = k0; k < k1; k += 4) {
        float2 q2 = *(const float2*)(qrow + k + half * 2);  // (Kh, Kh+1) of row n
        float2 t2 = *(const float2*)(trow + k + half * 2);  // (Kh, Kh+1) of tgt m
        v2f a; a.x = t2.x * tscale; a.y = t2.y * tscale;
        v2f b; b.x = q2.x;          b.y = q2.y;
        acc = __builtin_amdgcn_wmma_f32_16x16x4_f32(
            /*neg_a=*/false, a, /*neg_b=*/false, b,
            /*c_mod=*/(short)0, acc, /*reuse_a=*/false, /*reuse_b=*/false);
        sumsq = fmaf(q2.x, q2.x, sumsq);
        sumsq = fmaf(q2.y, q2.y, sumsq);
    }
    // combine lane l (K0,K1,...) with lane l+16 (K2,K3,...) for full row sumsq
    sumsq += __shfl_down(sumsq, 16, 32);

    if (lane < 16) {
        sqq[seg * BANK + nrow] = sumsq;
#pragma unroll
        for (int v = 0; v < 8; ++v)           // M=v == target b, N=lane -> nrow
            dots[((seg * 8 + v) * BANK) + nrow] = acc[v];
    }
}

// ---------------------------------------------------------------------------
// Kernel 2: per-b ||t||^2, d2 assembly, argmin over BANK (tie -> lowest idx).
// ---------------------------------------------------------------------------
__global__ __launch_bounds__(256) void apr_argmin(
    const float* __restrict__ tgt, const float* __restrict__ dots,
    const float* __restrict__ sqq, int* __restrict__ min_idx,
    float* __restrict__ closest_dist)
{
    const int b = blockIdx.x;
    const int tid = threadIdx.x;
    __shared__ float sval[256];
    __shared__ int   sidx[256];

    // sq_t
    float s = 0.0f;
    const float* tr = tgt + (size_t)b * DFEAT;
    for (int i = tid; i < DFEAT; i += 256) { float v = tr[i]; s = fmaf(v, v, s); }
    sval[tid] = s; __syncthreads();
    for (int off = 128; off > 0; off >>= 1) {
        if (tid < off) sval[tid] += sval[tid + off];
        __syncthreads();
    }
    const float sq_t = sval[0];
    __syncthreads();

    float best = INFINITY; int bestn = BANK;
    for (int n = tid; n < BANK; n += 256) {
        float dot = 0.0f, q = 0.0f;
#pragma unroll
        for (int seg = 0; seg < NSEG; ++seg) {
            dot += dots[(seg * 8 + b) * BANK + n];
            q   += sqq[seg * BANK + n];
        }
        float d2 = q - 2.0f * dot + sq_t;
        if (d2 < best || (d2 == best && n < bestn)) { best = d2; bestn = n; }
    }
    sval[tid] = best; sidx[tid] = bestn; __syncthreads();
    for (int off = 128; off > 0; off >>= 1) {
        if (tid < off) {
            float ov = sval[tid + off]; int oi = sidx[tid + off];
            if (ov < sval[tid] || (ov == sval[tid] && oi < sidx[tid])) {
                sval[tid] = ov; sidx[tid] = oi;
            }
        }
        __syncthreads();
    }
    if (tid == 0) {
        min_idx[b] = sidx[0];
        closest_dist[b] = sqrtf(fmaxf(sval[0], 1e-12f));
    }
}

// ---------------------------------------------------------------------------
// Kernel 3: squared distance of each augmented view to the closest bank row.
// One block per (b,a).
// ---------------------------------------------------------------------------
__global__ __launch_bounds__(256) void apr_daug(
    const float* __restrict__ queue, const float* __restrict__ aug,
    const int* __restrict__ min_idx, float* __restrict__ d_aug2)
{
    const int ba = blockIdx.x;               // b*NA + a
    const int b = ba >> 4;
    const int tid = threadIdx.x;
    __shared__ float sval[256];

    const float* __restrict__ cl = queue + (size_t)min_idx[b] * DFEAT;
    const float* __restrict__ ar = aug   + (size_t)ba * DFEAT;
    float s = 0.0f;
    for (int i = tid * 4; i < DFEAT; i += 256 * 4) {
        float4 a4 = *(const float4*)(ar + i);
        float4 c4 = *(const float4*)(cl + i);
        float dx = a4.x - c4.x, dy = a4.y - c4.y;
        float dz = a4.z - c4.z, dw = a4.w - c4.w;
        s = fmaf(dx, dx, s); s = fmaf(dy, dy, s);
        s = fmaf(dz, dz, s); s = fmaf(dw, dw, s);
    }
    sval[tid] = s; __syncthreads();
    for (int off = 128; off > 0; off >>= 1) {
        if (tid < off) sval[tid] += sval[tid + off];
        __syncthreads();
    }
    if (tid == 0) d_aug2[ba] = sval[0];
}

// ---------------------------------------------------------------------------
// Kernel 4: mask + stable top-5 smallest distances among A=16 (ties -> lowest
// index, matching lax.top_k), valid flags + count.
// ---------------------------------------------------------------------------
__global__ __launch_bounds__(32) void apr_select(
    const float* __restrict__ d_aug2, const float* __restrict__ closest_dist,
    int* __restrict__ sel_idx, int* __restrict__ sel_valid, int* __restrict__ count)
{
    const int b = threadIdx.x;
    if (b >= NB) return;
    const float cd = closest_dist[b];
    float md[NA]; int taken[NA];
    for (int a = 0; a < NA; ++a) {
        float da = sqrtf(fmaxf(d_aug2[b * NA + a], 1e-12f));
        md[a] = (da <= cd) ? da : INFINITY;
        taken[a] = 0;
    }
    int cnt = 0;
    for (int j = 0; j < KSEL; ++j) {
        float bv = INFINITY; int bi = -1;
        for (int a = 0; a < NA; ++a) {
            if (!taken[a] && (bi < 0 || md[a] < bv)) { bv = md[a]; bi = a; }
        }
        taken[bi] = 1;
        int v = (bv < INFINITY) ? 1 : 0;
        sel_idx[b * KSEL + j] = bi;
        sel_valid[b * KSEL + j] = v;
        cnt += v;
    }
    count[b] = cnt;
}

// ---------------------------------------------------------------------------
// Kernel 5: per-pixel softmax over C of selected augs, average (or pseudo
// label fallback), argmax -> refined. Coalesced in w.
// ---------------------------------------------------------------------------
__global__ __launch_bounds__(256) void apr_labels(
    const float* __restrict__ logits, const float* __restrict__ pseudo,
    const int* __restrict__ sel_idx, const int* __restrict__ sel_valid,
    const int* __restrict__ count, int* __restrict__ refined,
    float* __restrict__ soft_out)
{
    const int p = blockIdx.x * 256 + threadIdx.x;      // over B*HW
    if (p >= NB * HW) return;
    const int b = p >> 14;
    const int hw = p & (HW - 1);

    float acc[NC];
#pragma unroll
    for (int c = 0; c < NC; ++c) acc[c] = 0.0f;

    const int cnt = count[b];
    for (int j = 0; j < KSEL; ++j) {
        if (!sel_valid[b * KSEL + j]) continue;
        const int a = sel_idx[b * KSEL + j];
        const float* __restrict__ lp =
            logits + (size_t)(b * NA + a) * NC * HW + hw;
        float lv[NC];
        float mx = -INFINITY;
#pragma unroll
        for (int c = 0; c < NC; ++c) { lv[c] = lp[(size_t)c * HW]; mx = fmaxf(mx, lv[c]); }
        float ssum = 0.0f;
#pragma unroll
        for (int c = 0; c < NC; ++c) { lv[c] = expf(lv[c] - mx); ssum += lv[c]; }
        const float inv = 1.0f / ssum;
#pragma unroll
        for (int c = 0; c < NC; ++c) acc[c] = fmaf(lv[c], inv, acc[c]);
    }

    const float denom = 1.0f / (float)((cnt > 0) ? cnt : 1);
    int bestc = 0; float bestv = -INFINITY;
#pragma unroll
    for (int c = 0; c < NC; ++c) {
        float v = (cnt > 0) ? acc[c] * denom
                            : pseudo[(size_t)(b * NC + c) * HW + hw];
        soft_out[(size_t)(b * NC + c) * HW + hw] = v;
        if (v > bestv) { bestv = v; bestc = c; }   // first max wins (argmax)
    }
    refined[p] = bestc;
}

// ---------------------------------------------------------------------------
extern "C" void kernel_launch(void* const* d_in, const int* in_sizes, int n_in,
                              void* d_out, int out_size, void* d_ws, size_t ws_size,
                              hipStream_t stream) {
    const float* queue  = (const float*)d_in[0];   // [2048, 512,16,16]
    const float* tgt    = (const float*)d_in[1];   // [8, 512,16,16]
    const float* aug    = (const float*)d_in[2];   // [8,16, 512,16,16]
    const float* logits = (const float*)d_in[3];   // [8,16,19,128,128]
    const float* pseudo = (const float*)d_in[4];   // [8,19,128,128]
    (void)in_sizes; (void)n_in; (void)out_size; (void)ws_size;

    // Workspace layout (floats unless noted)
    float* ws = (float*)d_ws;
    float* dots   = ws;                          // NSEG*8*BANK  = 131072
    float* sqq    = dots + NSEG * 8 * BANK;      // NSEG*BANK    = 16384
    float* cdist  = sqq + NSEG * BANK;           // 8
    float* daug2  = cdist + NB;                  // 128
    int*   minidx = (int*)(daug2 + NB * NA);     // 8
    int*   selidx = minidx + NB;                 // 40
    int*   selval = selidx + NB * KSEL;          // 40
    int*   cnt    = selval + NB * KSEL;          // 8

    // Outputs: refined [8,128,128] int32 then soft_labels [8,19,128,128] f32
    int*   refined  = (int*)d_out;
    float* soft_out = (float*)d_out + NB * HW;

    apr_gemm_partial<<<NTILES * NSEG, 32, 0, stream>>>(queue, tgt, dots, sqq);
    apr_argmin<<<NB, 256, 0, stream>>>(tgt, dots, sqq, minidx, cdist);
    apr_daug<<<NB * NA, 256, 0, stream>>>(queue, aug, minidx, daug2);
    apr_select<<<1, 32, 0, stream>>>(daug2, cdist, selidx, selval, cnt);
    apr_labels<<<(NB * HW + 255) / 256, 256, 0, stream>>>(
        logits, pseudo, selidx, selval, cnt, refined, soft_out);
}